// RGAS_26268019982502
// MI455X (gfx1250) — compile-verified
//
#include <hip/hip_runtime.h>
#include <hip/hip_bf16.h>

typedef __attribute__((ext_vector_type(2))) float v2f;
typedef __attribute__((ext_vector_type(8))) float v8f;

#define HW   1024
#define INP  256
#define CS   64
#define AOUT 256
#define CF   513
#define C5   256

// ---------------------------------------------------------------------------
// V_WMMA_F32_16X16X4_F32 wrapper (wave32).
// A 16x4: lanes 0-15 hold K=k,k+1 in (x,y); lanes 16-31 hold K=k+2,k+3.
// B 4x16: reg0 lanes0-15 row k / lanes16-31 row k+2; reg1 rows k+1 / k+3.
// C/D:    reg r -> (M=r, N=lane) lanes 0-15, (M=r+8, N=lane-16) lanes 16-31.
// ---------------------------------------------------------------------------
__device__ inline v8f wmma4(v2f a, v2f b, v8f c) {
    return __builtin_amdgcn_wmma_f32_16x16x4_f32(
        /*neg_a=*/false, a, /*neg_b=*/false, b,
        /*c_mod=*/(short)0, c, /*reuse_a=*/false, /*reuse_b=*/false);
}

// ---------------------------------------------------------------------------
// Wave computes a 64x32 output strip: 4 M-tiles x 2 N-tiles, 8 accumulators.
// ---------------------------------------------------------------------------
template <typename FA, typename FB>
__device__ inline void gemm_strip42(v8f (&acc)[8], int K, FA ldA, FB ldB) {
    const int lane = threadIdx.x & 31;
    const int half = lane >> 4;
    const int idx  = lane & 15;
    for (int k = 0; k < K; k += 4) {
        const int k0 = k + 2 * half;
        v2f a0, a1, a2, a3, b0, b1;
        a0.x = ldA(idx,      k0); a0.y = ldA(idx,      k0 + 1);
        a1.x = ldA(16 + idx, k0); a1.y = ldA(16 + idx, k0 + 1);
        a2.x = ldA(32 + idx, k0); a2.y = ldA(32 + idx, k0 + 1);
        a3.x = ldA(48 + idx, k0); a3.y = ldA(48 + idx, k0 + 1);
        b0.x = ldB(k0, idx);      b0.y = ldB(k0 + 1, idx);
        b1.x = ldB(k0, 16 + idx); b1.y = ldB(k0 + 1, 16 + idx);
        acc[0] = wmma4(a0, b0, acc[0]); acc[1] = wmma4(a0, b1, acc[1]);
        acc[2] = wmma4(a1, b0, acc[2]); acc[3] = wmma4(a1, b1, acc[3]);
        acc[4] = wmma4(a2, b0, acc[4]); acc[5] = wmma4(a2, b1, acc[5]);
        acc[6] = wmma4(a3, b0, acc[6]); acc[7] = wmma4(a3, b1, acc[7]);
    }
}

// st(m in [0,64), n in [0,32), value)
template <typename FS>
__device__ inline void store_strip42(const v8f (&acc)[8], FS st) {
    const int lane = threadIdx.x & 31;
    const int half = lane >> 4;
    const int n    = lane & 15;
#pragma unroll
    for (int q = 0; q < 4; ++q)
#pragma unroll
        for (int p = 0; p < 2; ++p)
#pragma unroll
            for (int r = 0; r < 8; ++r)
                st(q * 16 + r + 8 * half, p * 16 + n, acc[q * 2 + p][r]);
}

// ---------------------------------------------------------------------------
// 32x32 strip: 2 M-tiles x 2 N-tiles, 4 accumulators.
// ---------------------------------------------------------------------------
template <typename FA, typename FB>
__device__ inline void gemm_strip22(v8f (&acc)[4], int K, FA ldA, FB ldB) {
    const int lane = threadIdx.x & 31;
    const int half = lane >> 4;
    const int idx  = lane & 15;
    for (int k = 0; k < K; k += 4) {
        const int k0 = k + 2 * half;
        v2f a0, a1, b0, b1;
        a0.x = ldA(idx,      k0); a0.y = ldA(idx,      k0 + 1);
        a1.x = ldA(16 + idx, k0); a1.y = ldA(16 + idx, k0 + 1);
        b0.x = ldB(k0, idx);      b0.y = ldB(k0 + 1, idx);
        b1.x = ldB(k0, 16 + idx); b1.y = ldB(k0 + 1, 16 + idx);
        acc[0] = wmma4(a0, b0, acc[0]); acc[1] = wmma4(a0, b1, acc[1]);
        acc[2] = wmma4(a1, b0, acc[2]); acc[3] = wmma4(a1, b1, acc[3]);
    }
}

// st(m in [0,32), n in [0,32), value)
template <typename FS>
__device__ inline void store_strip22(const v8f (&acc)[4], FS st) {
    const int lane = threadIdx.x & 31;
    const int half = lane >> 4;
    const int n    = lane & 15;
#pragma unroll
    for (int q = 0; q < 2; ++q)
#pragma unroll
        for (int p = 0; p < 2; ++p)
#pragma unroll
            for (int r = 0; r < 8; ++r)
                st(q * 16 + r + 8 * half, p * 16 + n, acc[q * 2 + p][r]);
}

__device__ inline float relu_(float v) { return v > 0.f ? v : 0.f; }

// ---------------------------------------------------------------------------
// k0: fold BN params into per-channel scale/bias, pack W123 = [w1;w2;w3].
// ---------------------------------------------------------------------------
__global__ void __launch_bounds__(256)
k0_prep(const float* w1, const float* b1, const float* g1, const float* bt1, const float* m1, const float* v1,
        const float* w2, const float* b2, const float* g2, const float* bt2, const float* m2, const float* v2,
        const float* w3, const float* b3, const float* g3, const float* bt3, const float* m3, const float* v3,
        const float* b5, const float* g5, const float* bt5, const float* m5, const float* v5,
        float* W123, float* sc, float* bi, float* s5f, float* t5f) {
    const int i = blockIdx.x * blockDim.x + threadIdx.x;
    if (i < 192 * 256) {
        const int row = i >> 8, col = i & 255;
        const float* w = row < 64 ? w1 : (row < 128 ? w2 : w3);
        W123[i] = w[(row & 63) * 256 + col];
    }
    if (i < 192) {
        const int set = i >> 6, c = i & 63;
        const float *g, *bt, *m, *v, *bb;
        if (set == 0)      { g = g1; bt = bt1; m = m1; v = v1; bb = b1; }
        else if (set == 1) { g = g2; bt = bt2; m = m2; v = v2; bb = b2; }
        else               { g = g3; bt = bt3; m = m3; v = v3; bb = b3; }
        const float s = g[c] * rsqrtf(v[c] + 1e-5f);
        sc[i] = s;
        bi[i] = (bb[c] - m[c]) * s + bt[c];
    }
    if (i < 256) {
        const float s = g5[i] * rsqrtf(v5[i] + 1e-5f);
        s5f[i] = s;
        t5f[i] = (b5[i] - m5[i]) * s + bt5[i];
    }
}

// ---------------------------------------------------------------------------
// k1: fused theta/phi/e3:  [192,256] x [256,1024] per batch, BN-fold + relu.
// ---------------------------------------------------------------------------
__global__ void __launch_bounds__(256)
k1_qkv(const float* __restrict__ x, const float* __restrict__ W123,
       const float* __restrict__ sc, const float* __restrict__ bi,
       float* __restrict__ theta, float* __restrict__ phi, float* __restrict__ e3, int b) {
    const int wave  = threadIdx.x >> 5;
    const int job   = blockIdx.x * 8 + wave;      // 0..95
    const int mQuad = job >> 5;                   // 0..2
    const int nBase = (job & 31) * 32;
    const int mBase = mQuad * 64;
    const float* xb = x + (size_t)b * INP * HW;
    float* dst = (mQuad == 0) ? theta : (mQuad == 1 ? phi : e3);

    auto ldA = [&](int m, int k) { return W123[(mBase + m) * 256 + k]; };
    auto ldB = [&](int k, int n) { return xb[k * HW + nBase + n]; };
    v8f acc[8] = {};
    gemm_strip42(acc, 256, ldA, ldB);

    store_strip42(acc, [&](int m, int n, float v) {
        const int gm = mBase + m;
        dst[m * HW + nBase + n] = relu_(v * sc[gm] + bi[gm]);
    });
}

// k1b: xe[p] = mean_c e3[c,p]
__global__ void __launch_bounds__(256)
k1b_xe(const float* __restrict__ e3, float* __restrict__ xe) {
    const int p = blockIdx.x * blockDim.x + threadIdx.x;
    if (p < HW) {
        float s = 0.f;
#pragma unroll 8
        for (int c = 0; c < CS; ++c) s += e3[c * HW + p];
        xe[p] = s * (1.f / 64.f);
    }
}

// ---------------------------------------------------------------------------
// k2: A1 = w4 * phi_r  [256,1024]x[1024,64];  A2 = theta * w4^T [64,1024]x[1024,256]
// phi_r is the raw phi buffer reinterpreted row-major as [1024,64].
// ---------------------------------------------------------------------------
__global__ void __launch_bounds__(256)
k2_factors(const float* __restrict__ w4, const float* __restrict__ theta,
           const float* __restrict__ phi, float* __restrict__ A1, float* __restrict__ A2) {
    const int wave = threadIdx.x >> 5;
    const int t = blockIdx.x * 8 + wave;          // 0..15
    if (t < 8) {                                   // A1: 4 mQuads x 2 nDoubles
        const int mBase = (t >> 1) * 64, nBase = (t & 1) * 32;
        auto ldA = [&](int m, int k) { return w4[(mBase + m) * HW + k]; };
        auto ldB = [&](int k, int n) { return phi[k * CS + nBase + n]; };
        v8f acc[8] = {};
        gemm_strip42(acc, HW, ldA, ldB);
        store_strip42(acc, [&](int m, int n, float v) {
            A1[(mBase + m) * CS + nBase + n] = v;
        });
    } else {                                       // A2: 1 mQuad x 8 nDoubles
        const int u = t - 8;
        const int mBase = 0, nBase = u * 32;
        auto ldA = [&](int m, int k) { return theta[(mBase + m) * HW + k]; };
        auto ldB = [&](int k, int n) { return w4[(nBase + n) * HW + k]; };
        v8f acc[8] = {};
        gemm_strip42(acc, HW, ldA, ldB);
        store_strip42(acc, [&](int m, int n, float v) {
            A2[(mBase + m) * AOUT + nBase + n] = v;
        });
    }
}

// ---------------------------------------------------------------------------
// k3: affA = relu(A1*theta + b4)   [256,64]x[64,1024] -> [256,1024]
//     affBT = relu(phi_r*A2 + b4)  [1024,64]x[64,256] -> [1024,256] (p-major)
// ---------------------------------------------------------------------------
__global__ void __launch_bounds__(256)
k3_aff(const float* __restrict__ A1, const float* __restrict__ A2,
       const float* __restrict__ theta, const float* __restrict__ phi,
       const float* __restrict__ b4, float* __restrict__ affA, float* __restrict__ affBT) {
    const int wave = threadIdx.x >> 5;
    const int t = blockIdx.x * 8 + wave;          // 0..255
    if (t < 128) {                                 // affA: 4 mQuads x 32 nDoubles
        const int mBase = (t >> 5) * 64, nBase = (t & 31) * 32;
        auto ldA = [&](int m, int k) { return A1[(mBase + m) * CS + k]; };
        auto ldB = [&](int k, int n) { return theta[k * HW + nBase + n]; };
        v8f acc[8] = {};
        gemm_strip42(acc, CS, ldA, ldB);
        store_strip42(acc, [&](int m, int n, float v) {
            affA[(mBase + m) * HW + nBase + n] = relu_(v + b4[mBase + m]);
        });
    } else {                                       // affBT: 16 mQuads x 8 nDoubles
        const int u = t - 128;
        const int mBase = (u >> 3) * 64, nBase = (u & 7) * 32;
        auto ldA = [&](int m, int k) { return phi[(mBase + m) * CS + k]; };
        auto ldB = [&](int k, int n) { return A2[k * AOUT + nBase + n]; };
        v8f acc[8] = {};
        gemm_strip42(acc, CS, ldA, ldB);
        store_strip42(acc, [&](int m, int n, float v) {
            affBT[(mBase + m) * AOUT + nBase + n] = relu_(v + b4[nBase + n]);
        });
    }
}

// ---------------------------------------------------------------------------
// k4: z = relu(s5 * (W5 . y) + t5), y = [xe; affA; affB], K = 1+256+256.
// One block per 32-column slab; both B tiles staged into LDS with
// GLOBAL_LOAD_ASYNC_TO_LDS_B128 (+ s_wait_asynccnt + barrier), then 8 waves
// each compute a 32x32 strip over K=512 reading B from LDS.
// LDS addresses are derived from the shared pointer via ptrtoint (low 32 bits
// of a generic LDS address == DS byte address, ISA 10.2) so the shared buffer
// escapes and the ds_loads cannot be undef-folded.
// affBT slab stored with 260-float pitch -> conflict-free k-strided ds reads.
// ---------------------------------------------------------------------------
#define AFFBT_PITCH 260                    // floats; 1040 B rows
#define LDS_AFFA_BYTES (256 * 32 * 4)      // 32 KB
#define LDS_TOTAL_BYTES (LDS_AFFA_BYTES + 32 * AFFBT_PITCH * 4)

__global__ void __launch_bounds__(256)
k4_z(const float* __restrict__ w5, const float* __restrict__ affA,
     const float* __restrict__ affBT, const float* __restrict__ xe,
     const float* __restrict__ s5f, const float* __restrict__ t5f, float* __restrict__ z) {
    __shared__ __align__(16) char smem[LDS_TOTAL_BYTES];
    const float* affA_s  = (const float*)smem;                       // [k][n], pitch 32
    const float* affBT_s = (const float*)(smem + LDS_AFFA_BYTES);    // [n][k], pitch 260

    const int nBase = blockIdx.x * 32;
    const int t     = threadIdx.x;

    // --- stage affA slab [256k x 32n]: 8 x (32 rows x 8 x 16B) async issues ---
#pragma unroll
    for (int i = 0; i < 8; ++i) {
        const int row = i * 32 + (t >> 3);           // k: 0..255
        const int seg = t & 7;                       // 16-byte segment in row
        const unsigned lds = (unsigned)(size_t)(smem + row * 128 + seg * 16);
        const float* g = affA + (size_t)row * HW + nBase + seg * 4;
        asm volatile("global_load_async_to_lds_b128 %0, %1, off"
                     :: "v"(lds), "v"(g) : "memory");
    }
    // --- stage affBT slab [32n x 256k] with 260-float pitch ---
#pragma unroll
    for (int i = 0; i < 8; ++i) {
        const int n   = i * 4 + (t >> 6);            // 0..31
        const int seg = t & 63;                      // 16-byte segment along k
        const unsigned lds =
            (unsigned)(size_t)(smem + LDS_AFFA_BYTES + n * (AFFBT_PITCH * 4) + seg * 16);
        const float* g = affBT + (size_t)(nBase + n) * AOUT + seg * 4;
        asm volatile("global_load_async_to_lds_b128 %0, %1, off"
                     :: "v"(lds), "v"(g) : "memory");
    }
    asm volatile("s_wait_asynccnt 0x0" ::: "memory");
    __syncthreads();

    // --- compute: wave w handles rows [w*32, w*32+32) ---
    const int wave  = t >> 5;
    const int mBase = wave * 32;

    auto ldA1 = [&](int m, int k) { return w5[(mBase + m) * CF + 1 + k]; };
    auto ldB1 = [&](int k, int n) { return affA_s[k * 32 + n]; };
    v8f acc[4] = {};
    gemm_strip22(acc, AOUT, ldA1, ldB1);

    auto ldA2 = [&](int m, int k) { return w5[(mBase + m) * CF + 1 + AOUT + k]; };
    auto ldB2 = [&](int k, int n) { return affBT_s[n * AFFBT_PITCH + k]; };
    gemm_strip22(acc, AOUT, ldA2, ldB2);

    store_strip22(acc, [&](int m, int n, float v) {
        const int gm = mBase + m, gn = nBase + n;
        const float g = v + w5[gm * CF] * xe[gn];
        z[gm * HW + gn] = relu_(g * s5f[gm] + t5f[gm]);
    });
}

// k5: a[p] = sigmoid(w6 . z[:,p] + b6)
__global__ void __launch_bounds__(256)
k5_gate(const float* __restrict__ z, const float* __restrict__ w6,
        const float* __restrict__ b6, float* __restrict__ avec) {
    const int p = blockIdx.x * blockDim.x + threadIdx.x;
    if (p < HW) {
        float s = b6[0];
#pragma unroll 8
        for (int o = 0; o < C5; ++o) s += w6[o] * z[o * HW + p];
        avec[p] = 1.f / (1.f + __expf(-s));
    }
}

// k6: out[b,c,p] = x[b,c,p] * a[p]
__global__ void __launch_bounds__(256)
k6_out(const float* __restrict__ x, const float* __restrict__ avec,
       float* __restrict__ out, int b) {
    const int i = blockIdx.x * blockDim.x + threadIdx.x;   // 256*1024
    if (i < INP * HW) {
        const size_t off = (size_t)b * INP * HW + i;
        out[off] = x[off] * avec[i & (HW - 1)];
    }
}

// ---------------------------------------------------------------------------
extern "C" void kernel_launch(void* const* d_in, const int* in_sizes, int n_in,
                              void* d_out, int out_size, void* d_ws, size_t ws_size,
                              hipStream_t stream) {
    const float* x   = (const float*)d_in[0];
    const float* w1  = (const float*)d_in[1];  const float* b1  = (const float*)d_in[2];
    const float* g1  = (const float*)d_in[3];  const float* bt1 = (const float*)d_in[4];
    const float* m1  = (const float*)d_in[5];  const float* v1  = (const float*)d_in[6];
    const float* w2  = (const float*)d_in[7];  const float* b2  = (const float*)d_in[8];
    const float* g2  = (const float*)d_in[9];  const float* bt2 = (const float*)d_in[10];
    const float* m2  = (const float*)d_in[11]; const float* v2  = (const float*)d_in[12];
    const float* w3  = (const float*)d_in[13]; const float* b3  = (const float*)d_in[14];
    const float* g3  = (const float*)d_in[15]; const float* bt3 = (const float*)d_in[16];
    const float* m3  = (const float*)d_in[17]; const float* v3  = (const float*)d_in[18];
    const float* w4  = (const float*)d_in[19]; const float* b4  = (const float*)d_in[20];
    const float* w5  = (const float*)d_in[21]; const float* b5  = (const float*)d_in[22];
    const float* g5  = (const float*)d_in[23]; const float* bt5 = (const float*)d_in[24];
    const float* m5  = (const float*)d_in[25]; const float* v5  = (const float*)d_in[26];
    const float* w6  = (const float*)d_in[27]; const float* b6  = (const float*)d_in[28];
    float* out = (float*)d_out;

    // workspace carve-up (floats)
    float* ws    = (float*)d_ws;
    float* W123  = ws;                  // 49152
    float* sc    = W123 + 49152;        // 192
    float* bi    = sc + 192;            // 192
    float* s5f   = bi + 192;            // 256
    float* t5f   = s5f + 256;           // 256
    float* xe    = t5f + 256;           // 1024
    float* theta = xe + 1024;           // 65536
    float* phi   = theta + 65536;       // 65536
    float* e3    = phi + 65536;         // 65536
    float* A1    = e3 + 65536;          // 16384
    float* A2    = A1 + 16384;          // 16384
    float* affA  = A2 + 16384;          // 262144
    float* affBT = affA + 262144;       // 262144
    float* zbuf  = affBT + 262144;      // 262144
    float* avec  = zbuf + 262144;       // 1024   (total ~4.3 MB)

    k0_prep<<<192, 256, 0, stream>>>(w1, b1, g1, bt1, m1, v1,
                                     w2, b2, g2, bt2, m2, v2,
                                     w3, b3, g3, bt3, m3, v3,
                                     b5, g5, bt5, m5, v5,
                                     W123, sc, bi, s5f, t5f);

    for (int b = 0; b < 32; ++b) {
        k1_qkv<<<12, 256, 0, stream>>>(x, W123, sc, bi, theta, phi, e3, b);
        k1b_xe<<<4, 256, 0, stream>>>(e3, xe);
        k2_factors<<<2, 256, 0, stream>>>(w4, theta, phi, A1, A2);
        k3_aff<<<32, 256, 0, stream>>>(A1, A2, theta, phi, b4, affA, affBT);
        k4_z<<<32, 256, 0, stream>>>(w5, affA, affBT, xe, s5f, t5f, zbuf);
        k5_gate<<<4, 256, 0, stream>>>(zbuf, w6, b6, avec);
        k6_out<<<1024, 256, 0, stream>>>(x, avec, out, b);
    }
}